// PeriodicKNN_PyG_76347338654306
// MI455X (gfx1250) — compile-verified
//
#include <hip/hip_runtime.h>
#include <hip/hip_bf16.h>

// Periodic KNN (N=8192, k=19, 27 images) for MI455X / gfx1250.
//
// Strategy:
//   pre:   pos = frac@cell, pa2=|p|^2 ; B[s][j] = (-2(p_j+sv_s), |p_j+sv_s|^2)  (3.4MB, L2-resident)
//   main:  per 16x16 tile, per shift:  g = A x B  via V_WMMA_F32_16X16X4_F32 with
//          A row = [px,py,pz,1]; running min over 27 shifts; |p_i|^2 dropped (row-constant,
//          order preserving) -> C = 0.  Transpose tile via LDS, per-lane register top-19.
//   merge: 4-way merge of column-partition partials, write edge_index + dist.

#define N_ATOMS 8192
#define KNN     19
#define NPART   4
#define COLS_PER_PART (N_ATOMS / NPART)          // 2048
#define NCOLT_PER_PART (COLS_PER_PART / 16)      // 128

typedef __attribute__((ext_vector_type(2))) float v2f;
typedef __attribute__((ext_vector_type(8))) float v8f;

// ---- workspace layout (bytes) ----
#define OFF_POS4 ((size_t)0)
#define OFF_B    (OFF_POS4 + (size_t)N_ATOMS * 4 * sizeof(float))                  // 131072
#define OFF_PD   (OFF_B    + (size_t)27 * N_ATOMS * 4 * sizeof(float))             // +3538944
#define OFF_PJ   (OFF_PD   + (size_t)NPART * N_ATOMS * KNN * sizeof(float))        // +2490368
// total ~8.65 MB

// ---------------------------------------------------------------------------
// Kernel 1: positions + shifted B operand
// ---------------------------------------------------------------------------
__global__ void knn_precompute(const float* __restrict__ frac,
                               const float* __restrict__ cell,
                               float* __restrict__ pos4,
                               float* __restrict__ B)
{
    const int j = blockIdx.x * blockDim.x + threadIdx.x;
    if (j >= N_ATOMS) return;

    const float fx = frac[j * 3 + 0];
    const float fy = frac[j * 3 + 1];
    const float fz = frac[j * 3 + 2];

    float c[9];
#pragma unroll
    for (int i = 0; i < 9; ++i) c[i] = cell[i];

    // pos = frac @ cell  (pos_c = sum_r frac_r * cell[r][c])
    const float px = fx * c[0] + fy * c[3] + fz * c[6];
    const float py = fx * c[1] + fy * c[4] + fz * c[7];
    const float pz = fx * c[2] + fy * c[5] + fz * c[8];
    const float pa2 = px * px + py * py + pz * pz;

    pos4[j * 4 + 0] = px;
    pos4[j * 4 + 1] = py;
    pos4[j * 4 + 2] = pz;
    pos4[j * 4 + 3] = pa2;

#pragma unroll
    for (int s = 0; s < 27; ++s) {
        const float sx = (float)(s / 9 - 1);
        const float sy = (float)((s / 3) % 3 - 1);
        const float sz = (float)(s % 3 - 1);
        const float svx = sx * c[0] + sy * c[3] + sz * c[6];
        const float svy = sx * c[1] + sy * c[4] + sz * c[7];
        const float svz = sx * c[2] + sy * c[5] + sz * c[8];
        const float qx = px + svx, qy = py + svy, qz = pz + svz;
        const float q2 = qx * qx + qy * qy + qz * qz;
        float* o = B + ((size_t)s * N_ATOMS + j) * 4;
        o[0] = -2.0f * qx;
        o[1] = -2.0f * qy;
        o[2] = -2.0f * qz;
        o[3] = q2;
    }
}

// ---------------------------------------------------------------------------
// Kernel 2: WMMA distance tiles + min over shifts + per-lane top-19
// ---------------------------------------------------------------------------
__global__ __launch_bounds__(256) void knn_main(const float* __restrict__ pos4,
                                                const float* __restrict__ Bbuf,
                                                float* __restrict__ pd,
                                                int* __restrict__ pj)
{
    __shared__ float tile[8][32][17];   // per-wave 32x16 tile, stride 17 -> conflict free

    const int lane = threadIdx.x & 31;
    const int w    = threadIdx.x >> 5;
    const int half = lane >> 4;          // 0: K0/K1 lanes, 1: K2/K3 lanes
    const int lc   = lane & 15;
    const int rowbase = (blockIdx.x * 8 + w) * 32;
    const int r    = rowbase + lane;     // row owned by this lane for selection
    const int part = blockIdx.y;

    // A tiles: A[m] = [px,py,pz,1]; VGPR0=K0|K2, VGPR1=K1|K3 per lane-half, M=lane&15
    const float4 pA0 = ((const float4*)pos4)[rowbase + lc];
    const float4 pA1 = ((const float4*)pos4)[rowbase + 16 + lc];
    v2f a0, a1;
    a0.x = half ? pA0.z : pA0.x;  a0.y = half ? 1.0f : pA0.y;
    a1.x = half ? pA1.z : pA1.x;  a1.y = half ? 1.0f : pA1.y;

    // private sorted top-19 (ascending), static indexing only -> stays in VGPRs
    float bd[KNN];
    int   bj[KNN];
#pragma unroll
    for (int t = 0; t < KNN; ++t) { bd[t] = __builtin_inff(); bj[t] = 0; }

    const v8f cz = {};  // C = 0 (|p|^2 dropped: row-constant, order preserving)

    for (int ct = 0; ct < NCOLT_PER_PART; ++ct) {
        const int col0 = part * COLS_PER_PART + ct * 16;
        const int j    = col0 + lc;
        // lane fetches its (K0,K1) or (K2,K3) pair of column j's float4 per shift
        const float* bp = Bbuf + (size_t)j * 4 + half * 2;

        v8f acc0, acc1;
#pragma unroll
        for (int v = 0; v < 8; ++v) { acc0[v] = 3.0e38f; acc1[v] = 3.0e38f; }

#pragma unroll 3
        for (int s = 0; s < 27; ++s) {
            const float2 bl = *(const float2*)(bp + (size_t)s * N_ATOMS * 4);
            v2f b; b.x = bl.x; b.y = bl.y;
            // g = A x B + 0  ->  |q|^2 - 2 p.q   (two row tiles share every B load)
            v8f g0 = __builtin_amdgcn_wmma_f32_16x16x4_f32(
                false, a0, false, b, (short)0, cz, false, false);
            v8f g1 = __builtin_amdgcn_wmma_f32_16x16x4_f32(
                false, a1, false, b, (short)0, cz, false, false);
#pragma unroll
            for (int v = 0; v < 8; ++v) {
                acc0[v] = __builtin_fminf(acc0[v], g0[v]);
                acc1[v] = __builtin_fminf(acc1[v], g1[v]);
            }
        }

        if (ct + 1 < NCOLT_PER_PART)
            __builtin_prefetch(bp + 64, 0, 1);   // next column tile, shift 0

        // D layout: VGPR v holds (M = v + 8*half, N = lc)  -> transpose via LDS
#pragma unroll
        for (int v = 0; v < 8; ++v) {
            tile[w][half * 8 + v][lc]      = acc0[v];
            tile[w][16 + half * 8 + v][lc] = acc1[v];
        }
        __syncthreads();

        // each lane scans its own row; first compare is the hot path
#pragma unroll
        for (int c = 0; c < 16; ++c) {
            const float g  = tile[w][lane][c];
            const int   jg = col0 + c;
            if (jg != r && g < bd[KNN - 1]) {
                bd[KNN - 1] = g; bj[KNN - 1] = jg;
#pragma unroll
                for (int t = KNN - 1; t > 0; --t) {
                    if (bd[t] < bd[t - 1]) {
                        float td = bd[t]; bd[t] = bd[t - 1]; bd[t - 1] = td;
                        int   tj = bj[t]; bj[t] = bj[t - 1]; bj[t - 1] = tj;
                    }
                }
            }
        }
        __syncthreads();
    }

    float* od = pd + ((size_t)part * N_ATOMS + r) * KNN;
    int*   oj = pj + ((size_t)part * N_ATOMS + r) * KNN;
#pragma unroll
    for (int t = 0; t < KNN; ++t) { od[t] = bd[t]; oj[t] = bj[t]; }
}

// ---------------------------------------------------------------------------
// Kernel 3: merge the NPART sorted partials, emit edge_index + dist
// ---------------------------------------------------------------------------
__global__ void knn_merge(const float* __restrict__ pos4,
                          const float* __restrict__ pd,
                          const int* __restrict__ pj,
                          int* __restrict__ oi,
                          float* __restrict__ of)
{
    const int r = blockIdx.x * blockDim.x + threadIdx.x;
    if (r >= N_ATOMS) return;

    const float pa2 = pos4[r * 4 + 3];
    const size_t NK = (size_t)N_ATOMS * KNN;

    const float* p0 = pd + ((size_t)0 * N_ATOMS + r) * KNN;
    const float* p1 = pd + ((size_t)1 * N_ATOMS + r) * KNN;
    const float* p2 = pd + ((size_t)2 * N_ATOMS + r) * KNN;
    const float* p3 = pd + ((size_t)3 * N_ATOMS + r) * KNN;
    const int*   q0 = pj + ((size_t)0 * N_ATOMS + r) * KNN;
    const int*   q1 = pj + ((size_t)1 * N_ATOMS + r) * KNN;
    const int*   q2 = pj + ((size_t)2 * N_ATOMS + r) * KNN;
    const int*   q3 = pj + ((size_t)3 * N_ATOMS + r) * KNN;

    int i0 = 0, i1 = 0, i2 = 0, i3 = 0;
    for (int t = 0; t < KNN; ++t) {
        const float v0 = (i0 < KNN) ? p0[i0] : __builtin_inff();
        const float v1 = (i1 < KNN) ? p1[i1] : __builtin_inff();
        const float v2 = (i2 < KNN) ? p2[i2] : __builtin_inff();
        const float v3 = (i3 < KNN) ? p3[i3] : __builtin_inff();
        float best = v0; int bp = 0;
        if (v1 < best) { best = v1; bp = 1; }
        if (v2 < best) { best = v2; bp = 2; }
        if (v3 < best) { best = v3; bp = 3; }
        int jn;
        if      (bp == 0) jn = q0[i0++];
        else if (bp == 1) jn = q1[i1++];
        else if (bp == 2) jn = q2[i2++];
        else              jn = q3[i3++];

        oi[(size_t)r * KNN + t]          = r;                       // edge_index row 0 (src)
        oi[NK + (size_t)r * KNN + t]     = jn;                      // edge_index row 1 (nbr)
        of[2 * NK + (size_t)r * KNN + t] =
            __builtin_sqrtf(__builtin_fmaxf(best + pa2, 0.0f));     // dist
    }
}

// ---------------------------------------------------------------------------
extern "C" void kernel_launch(void* const* d_in, const int* in_sizes, int n_in,
                              void* d_out, int out_size, void* d_ws, size_t ws_size,
                              hipStream_t stream)
{
    const float* frac = (const float*)d_in[0];   // [8192,3] f32
    const float* cell = (const float*)d_in[1];   // [3,3]    f32

    char*  ws   = (char*)d_ws;
    float* pos4 = (float*)(ws + OFF_POS4);
    float* B    = (float*)(ws + OFF_B);
    float* pd   = (float*)(ws + OFF_PD);
    int*   pj   = (int*)  (ws + OFF_PJ);

    knn_precompute<<<N_ATOMS / 256, 256, 0, stream>>>(frac, cell, pos4, B);
    knn_main<<<dim3(N_ATOMS / 256, NPART), 256, 0, stream>>>(pos4, B, pd, pj);
    knn_merge<<<N_ATOMS / 256, 256, 0, stream>>>(pos4, pd, pj, (int*)d_out, (float*)d_out);
}